// KAVTC_Module_G2_44092134260978
// MI455X (gfx1250) — compile-verified
//
#include <hip/hip_runtime.h>
#include <hip/hip_bf16.h>
#include <stdint.h>

// Problem constants (match reference)
#define BB 32
#define NN 4096
#define DD 512
#define MAXK 2048

typedef float v2f  __attribute__((ext_vector_type(2)));
typedef float v8f  __attribute__((ext_vector_type(8)));

// ---------------------------------------------------------------------------
// Kernel 1: text_q = text @ Wq^T + bq            (M=32, N=512, K=512, f32)
// One wave per 16x16 tile, V_WMMA_F32_16X16X4_F32 over K.
// A[16x4] f32 layout: lanes 0-15 hold {K=0,K=1}, lanes 16-31 hold {K=2,K=3},
// row M = lane&15, one K per VGPR -> contiguous float2 load per lane.
// B[4x16] = Wq^T, i.e. B[k][n] = Wq[n*D + k]  -> same contiguous pattern.
// ---------------------------------------------------------------------------
__global__ void gemm_textq_kernel(const float* __restrict__ text,
                                  const float* __restrict__ Wq,
                                  const float* __restrict__ bq,
                                  float* __restrict__ textq) {
  const int gtid = blockIdx.x * blockDim.x + threadIdx.x;
  const int wave = gtid >> 5;
  const int lane = threadIdx.x & 31;
  const int mtile = wave & 1;          // 32 rows -> 2 tiles
  const int ntile = wave >> 1;         // 512 cols -> 32 tiles
  const int l16   = lane & 15;
  const int khalf = (lane < 16) ? 0 : 2;

  const float* arow = text + (mtile * 16 + l16) * DD + khalf;
  const float* brow = Wq   + (ntile * 16 + l16) * DD + khalf;   // B = Wq^T

  v8f acc = {};
  for (int k0 = 0; k0 < DD; k0 += 4) {
    v2f a = *reinterpret_cast<const v2f*>(arow + k0);
    v2f b = *reinterpret_cast<const v2f*>(brow + k0);
    acc = __builtin_amdgcn_wmma_f32_16x16x4_f32(false, a, false, b,
                                                (short)0, acc, false, false);
  }
  const int col = ntile * 16 + l16;
  const float bias = bq[col];
  const int mbase = mtile * 16 + ((lane < 16) ? 0 : 8);
#pragma unroll
  for (int i = 0; i < 8; ++i)
    textq[(mbase + i) * DD + col] = acc[i] + bias;
}

// ---------------------------------------------------------------------------
// Kernel 2: v = text_q @ Wk                      (M=32, N=512, K=512, f32)
// B[k][n] = Wk[k*D + n]  (no transpose) -> two strided scalar loads per lane.
// ---------------------------------------------------------------------------
__global__ void gemm_v_kernel(const float* __restrict__ textq,
                              const float* __restrict__ Wk,
                              float* __restrict__ vvec) {
  const int gtid = blockIdx.x * blockDim.x + threadIdx.x;
  const int wave = gtid >> 5;
  const int lane = threadIdx.x & 31;
  const int mtile = wave & 1;
  const int ntile = wave >> 1;
  const int l16   = lane & 15;
  const int khalf = (lane < 16) ? 0 : 2;
  const int col   = ntile * 16 + l16;

  const float* arow = textq + (mtile * 16 + l16) * DD + khalf;

  v8f acc = {};
  for (int k0 = 0; k0 < DD; k0 += 4) {
    v2f a = *reinterpret_cast<const v2f*>(arow + k0);
    v2f b;
    b.x = Wk[(k0 + khalf + 0) * DD + col];
    b.y = Wk[(k0 + khalf + 1) * DD + col];
    acc = __builtin_amdgcn_wmma_f32_16x16x4_f32(false, a, false, b,
                                                (short)0, acc, false, false);
  }
  const int mbase = mtile * 16 + ((lane < 16) ? 0 : 8);
#pragma unroll
  for (int i = 0; i < 8; ++i)
    vvec[(mbase + i) * DD + col] = acc[i];
}

// ---------------------------------------------------------------------------
// Kernel 3: per-bag num_patches -> k_per_bag, and c[b] = text_q[b]·bk
// One block (256 threads) per bag. mask: jnp.bool_ -> 1 byte per element.
// ---------------------------------------------------------------------------
__global__ void perbag_kernel(const uint8_t* __restrict__ mask,
                              const float* __restrict__ textq,
                              const float* __restrict__ bk,
                              int* __restrict__ kpb,
                              float* __restrict__ cvec) {
  __shared__ int   scnt[256];
  __shared__ float sdot[256];
  const int b = blockIdx.x;
  const int t = threadIdx.x;

  int cnt = 0;
  for (int n = t; n < NN; n += 256) cnt += mask[b * NN + n] ? 0 : 1;
  float dot = 0.f;
  for (int d = t; d < DD; d += 256) dot += textq[b * DD + d] * bk[d];
  scnt[t] = cnt; sdot[t] = dot;
  __syncthreads();
  for (int s = 128; s > 0; s >>= 1) {
    if (t < s) { scnt[t] += scnt[t + s]; sdot[t] += sdot[t + s]; }
    __syncthreads();
  }
  if (t == 0) {
    int np = scnt[0];
    int k  = (int)((float)np * 0.8f);
    k = k < 1 ? 1 : (k > MAXK ? MAXK : k);
    k = k < np ? k : np;
    if (k == 0) k = 1;
    kpb[b]  = k;
    cvec[b] = sdot[0];
  }
}

// ---------------------------------------------------------------------------
// Kernel 4: scores[b,n] = bag[b,n]·v[b] + c[b], masked -> -inf
// One wave per row; lane reads 4 float4s (512 f32 / row), shuffle-reduce.
// ---------------------------------------------------------------------------
__global__ void scores_kernel(const float* __restrict__ bag,
                              const uint8_t* __restrict__ mask,
                              const float* __restrict__ vvec,
                              const float* __restrict__ cvec,
                              float* __restrict__ scores) {
  const int gtid = blockIdx.x * blockDim.x + threadIdx.x;
  const int row  = gtid >> 5;                 // 0 .. B*N-1
  const int lane = threadIdx.x & 31;
  const int b = row >> 12;                    // row / 4096
  const int n = row & (NN - 1);

  if (mask[b * NN + n]) {                     // wave-uniform branch
    if (lane == 0) scores[row] = -__builtin_huge_valf();
    return;
  }
  const float4* r = reinterpret_cast<const float4*>(bag + (size_t)row * DD);
  const float4* v = reinterpret_cast<const float4*>(vvec + b * DD);
  float acc = 0.f;
#pragma unroll
  for (int i = 0; i < 4; ++i) {
    const int f = lane + 32 * i;
    float4 x = r[f], y = v[f];
    acc += x.x * y.x + x.y * y.y + x.z * y.z + x.w * y.w;
  }
#pragma unroll
  for (int m = 16; m >= 1; m >>= 1) acc += __shfl_xor(acc, m, 32);
  if (lane == 0) scores[row] = acc + cvec[b];
}

// ---------------------------------------------------------------------------
// Kernel 5: per-bag bitonic sort of 4096 (score,idx), descending score,
// ascending index on ties (matches jax.lax.top_k ordering). 512 threads/bag.
// ---------------------------------------------------------------------------
__global__ void topk_sort_kernel(const float* __restrict__ scores,
                                 int* __restrict__ topidx) {
  __shared__ float s[NN];
  __shared__ int   id[NN];
  const int b = blockIdx.x;
  const int t = threadIdx.x;
  const int T = 512;

  for (int n = t; n < NN; n += T) { s[n] = scores[b * NN + n]; id[n] = n; }
  __syncthreads();

  for (int k = 2; k <= NN; k <<= 1) {
    for (int j = k >> 1; j > 0; j >>= 1) {
      for (int p = t; p < NN / 2; p += T) {
        const int i   = ((p & ~(j - 1)) << 1) | (p & (j - 1));
        const int ixj = i + j;
        const bool dirDesc = ((i & k) == 0);
        const float si = s[i], sj = s[ixj];
        const int   ii = id[i], ij = id[ixj];
        const bool iBefore = (si > sj) || (si == sj && ii < ij);
        if (dirDesc ? !iBefore : iBefore) {
          s[i] = sj; s[ixj] = si;
          id[i] = ij; id[ixj] = ii;
        }
      }
      __syncthreads();
    }
  }
  for (int m = t; m < MAXK; m += T) topidx[b * MAXK + m] = id[m];
}

// ---------------------------------------------------------------------------
// Kernel 6: gather top rows into d_out and emit new mask (as 0/1 float).
// One wave per output row (2 KB copy via float4), lane 0 writes mask bit.
// ---------------------------------------------------------------------------
__global__ void gather_kernel(const float* __restrict__ bag,
                              const int* __restrict__ topidx,
                              const int* __restrict__ kpb,
                              float* __restrict__ out,
                              float* __restrict__ outmask) {
  const int gtid = blockIdx.x * blockDim.x + threadIdx.x;
  const int row  = gtid >> 5;                 // 0 .. B*MAXK-1
  const int lane = threadIdx.x & 31;
  const int b = row >> 11;                    // row / 2048
  const int m = row & (MAXK - 1);

  const int src_n = topidx[row];
  const float4* src = reinterpret_cast<const float4*>(
      bag + ((size_t)b * NN + src_n) * DD);
  float4* dst = reinterpret_cast<float4*>(out + (size_t)row * DD);
#pragma unroll
  for (int i = 0; i < 4; ++i) {
    const int f = lane + 32 * i;
    dst[f] = src[f];
  }
  if (lane == 0) outmask[row] = (m >= kpb[b]) ? 1.0f : 0.0f;
}

// ---------------------------------------------------------------------------
extern "C" void kernel_launch(void* const* d_in, const int* in_sizes, int n_in,
                              void* d_out, int out_size, void* d_ws, size_t ws_size,
                              hipStream_t stream) {
  const float*   bag   = (const float*)d_in[0];            // [B,N,D]
  const uint8_t* mask  = (const uint8_t*)d_in[1];          // [B,N] bool (1B)
  const float*   text  = (const float*)d_in[2];            // [B,D]
  const float*   Wq    = (const float*)d_in[3];            // [D,D]
  const float*   bq    = (const float*)d_in[4];            // [D]
  const float*   Wk    = (const float*)d_in[5];            // [D,D]
  const float*   bk    = (const float*)d_in[6];            // [D]

  float* out     = (float*)d_out;                          // [B,MAXK,D]
  float* outmask = out + (size_t)BB * MAXK * DD;           // [B,MAXK] as f32

  // workspace layout
  float* textq  = (float*)d_ws;                            // B*D
  float* vvec   = textq + BB * DD;                         // B*D
  float* cvec   = vvec + BB * DD;                          // B
  int*   kpb    = (int*)(cvec + BB);                       // B
  float* scores = (float*)(kpb + BB);                      // B*N
  int*   topidx = (int*)(scores + BB * NN);                // B*MAXK

  // 1) text_q = text @ Wq^T + bq   (64 wmma tiles -> 8 blocks x 8 waves)
  gemm_textq_kernel<<<8, 256, 0, stream>>>(text, Wq, bq, textq);
  // 2) k_per_bag and c[b]
  perbag_kernel<<<BB, 256, 0, stream>>>(mask, textq, bk, kpb, cvec);
  // 3) v = text_q @ Wk
  gemm_v_kernel<<<8, 256, 0, stream>>>(textq, Wk, vvec);
  // 4) scores (one wave per row; B*N rows, 8 rows per 256-thread block)
  scores_kernel<<<(BB * NN) / 8, 256, 0, stream>>>(bag, mask, vvec, cvec, scores);
  // 5) per-bag bitonic top-k sort
  topk_sort_kernel<<<BB, 512, 0, stream>>>(scores, topidx);
  // 6) gather + new mask (one wave per output row)
  gather_kernel<<<(BB * MAXK) / 8, 256, 0, stream>>>(bag, topidx, kpb, out, outmask);
}